// DetectionLayer_31662498906496
// MI455X (gfx1250) — compile-verified
//
#include <hip/hip_runtime.h>
#include <hip/hip_bf16.h>
#include <stdint.h>

// Problem constants (match reference)
#define BB 8
#define NN 2000
#define CC 81
#define META_W (12 + CC)   // 93
#define MAXDET 100
#define SORT_N 2048
#define MIN_CONF 0.7f
#define NMS_TH 0.3f

// -------------------------------------------------------------------------
// Kernel 1: one wave32 per ROI.
//   - coalesced read of 81 probs, wave-reduce argmax (stable: lowest class id)
//   - gather only the winning class's 4 deltas (16 B instead of 1296 B/ROI)
//   - refine + clip, write box/score_key/cid to workspace
// -------------------------------------------------------------------------
__global__ __launch_bounds__(256) void detlayer_refine(
    const float* __restrict__ rois,     // (B,N,4)
    const float* __restrict__ probs,    // (B,N,C)
    const float* __restrict__ deltas,   // (B,N,C,4)
    const float* __restrict__ meta,     // (B,93)
    float4* __restrict__ o_box,         // (B,N)
    float*  __restrict__ o_score,       // (B,N)  (-1 if invalid)
    int*    __restrict__ o_cid)         // (B,N)
{
    int wave = blockIdx.x * (blockDim.x >> 5) + (threadIdx.x >> 5);
    int lane = threadIdx.x & 31;
    if (wave >= BB * NN) return;
    int b = wave / NN;
    int r = wave - b * NN;

    const float* p = probs + ((size_t)b * NN + r) * CC;
    // CDNA5 prefetch of the next ROI's prob row (global_prefetch_b8)
    __builtin_prefetch(p + CC, 0, 0);

    float bestv = -1.0f;
    int   besti = 0;
    for (int k = lane; k < CC; k += 32) {
        float v = p[k];
        if (v > bestv) { bestv = v; besti = k; }   // ascending k keeps first max
    }
    // wave32 butterfly reduction; tie -> lower class index (matches jnp.argmax)
    for (int off = 16; off > 0; off >>= 1) {
        float ov = __shfl_xor(bestv, off, 32);
        int   oi = __shfl_xor(besti, off, 32);
        if (ov > bestv || (ov == bestv && oi < besti)) { bestv = ov; besti = oi; }
    }

    if (lane == 0) {
        size_t rr = (size_t)b * NN + r;
        const float4 roi = *(const float4*)(rois + rr * 4);
        const float* d = deltas + (rr * CC + besti) * 4;
        float dy = d[0] * 0.1f, dx = d[1] * 0.1f;
        float dh = d[2] * 0.2f, dw = d[3] * 0.2f;

        float h = roi.z - roi.x;
        float w = roi.w - roi.y;
        float cy = roi.x + 0.5f * h + dy * h;
        float cx = roi.y + 0.5f * w + dx * w;
        h *= expf(dh);
        w *= expf(dw);
        float y1 = cy - 0.5f * h, x1 = cx - 0.5f * w;
        float y2 = y1 + h,        x2 = x1 + w;

        // window = (window_px - [0,0,1,1]) / ([h,w,h,w]-1)
        float ih = meta[4] - 1.0f;
        float iw = meta[5] - 1.0f;
        const float* wp = meta + (size_t)b * META_W + 7;
        float wy1 = wp[0] / ih, wx1 = wp[1] / iw;
        float wy2 = (wp[2] - 1.0f) / ih, wx2 = (wp[3] - 1.0f) / iw;
        y1 = fminf(fmaxf(y1, wy1), wy2);
        y2 = fminf(fmaxf(y2, wy1), wy2);
        x1 = fminf(fmaxf(x1, wx1), wx2);
        x2 = fminf(fmaxf(x2, wx1), wx2);

        bool valid = (besti > 0) && (bestv >= MIN_CONF);
        o_box[rr]   = make_float4(y1, x1, y2, x2);
        o_score[rr] = valid ? bestv : -1.0f;   // invalid sorts after all valid
        o_cid[rr]   = besti;
    }
}

// -------------------------------------------------------------------------
// Kernel 2: one 1024-thread workgroup per batch, everything in LDS.
//   bitonic sort (2048, stable desc) -> async global->LDS sorted gather ->
//   exact greedy NMS -> emit top MAXDET rows.
// -------------------------------------------------------------------------
__global__ __launch_bounds__(1024) void detlayer_sort_nms(
    const float4* __restrict__ i_box,
    const float*  __restrict__ i_score,
    const int*    __restrict__ i_cid,
    float* __restrict__ out)            // (B, MAXDET, 6)
{
    __shared__ unsigned long long s_key[SORT_N];   // 16 KB
    __shared__ float4 s_box[SORT_N];               // 32 KB
    __shared__ float  s_score[SORT_N];             //  8 KB
    __shared__ int    s_cid[SORT_N];               //  8 KB
    __shared__ int    s_removed[SORT_N];           //  8 KB
    __shared__ int    s_kept[MAXDET];
    __shared__ int    s_cnt[2];                    // [0]=valid count, [1]=kept count

    const int b   = blockIdx.x;
    const int tid = threadIdx.x;

    // ---- build sort keys: key = monotone(score) << 32 | ~origIndex ----
    // descending sort => higher score first; ties => lower index first (stable)
    for (int e = tid; e < SORT_N; e += 1024) {
        unsigned long long kk = 0ull;              // pads sort last
        if (e < NN) {
            float f = i_score[(size_t)b * NN + e];
            unsigned u = __float_as_uint(f);
            unsigned mono = (f >= 0.0f) ? (u | 0x80000000u) : ~u;
            kk = ((unsigned long long)mono << 32) | (unsigned)(~e);
        }
        s_key[e] = kk;
    }
    if (tid == 0) { s_cnt[0] = 0; s_cnt[1] = 0; }

    // ---- bitonic sort, descending ----
    for (int k = 2; k <= SORT_N; k <<= 1) {
        for (int j = k >> 1; j > 0; j >>= 1) {
            __syncthreads();
            for (int e = tid; e < SORT_N; e += 1024) {
                int x = e ^ j;
                if (x > e) {
                    unsigned long long a = s_key[e];
                    unsigned long long c = s_key[x];
                    bool sw = ((e & k) == 0) ? (a < c) : (a > c);  // descending
                    if (sw) { s_key[e] = c; s_key[x] = a; }
                }
            }
        }
    }
    __syncthreads();

    // ---- gather boxes/scores/cids into LDS in sorted order ----
    // CDNA5 async global->LDS copies (ASYNCcnt), per-lane scatter addresses.
    for (int e = tid; e < SORT_N; e += 1024) {
        s_removed[e] = 0;
        if (e < NN) {
            int idx = (int)(~(unsigned)s_key[e]);          // low 32 bits = ~index
            size_t g = (size_t)b * NN + idx;
            unsigned lds_box = (unsigned)(size_t)&s_box[e];
            unsigned lds_sc  = (unsigned)(size_t)&s_score[e];
            unsigned lds_cd  = (unsigned)(size_t)&s_cid[e];
            const void* gb = (const void*)&i_box[g];
            const void* gs = (const void*)&i_score[g];
            const void* gc = (const void*)&i_cid[g];
            asm volatile("global_load_async_to_lds_b128 %0, %1, off"
                         :: "v"(lds_box), "v"(gb) : "memory");
            asm volatile("global_load_async_to_lds_b32 %0, %1, off"
                         :: "v"(lds_sc), "v"(gs) : "memory");
            asm volatile("global_load_async_to_lds_b32 %0, %1, off"
                         :: "v"(lds_cd), "v"(gc) : "memory");
        } else {
            s_score[e] = -2.0f;                            // pads never valid
            s_cid[e]   = 0;
            s_box[e]   = make_float4(0.f, 0.f, 0.f, 0.f);
        }
    }
    asm volatile("s_wait_asynccnt 0" ::: "memory");
    __syncthreads();

    // ---- count valid (valid entries form a sorted prefix) ----
    int lv = 0;
    for (int e = tid; e < SORT_N; e += 1024)
        if (s_score[e] >= MIN_CONF) ++lv;
    if (lv) atomicAdd(&s_cnt[0], lv);
    __syncthreads();
    const int V = s_cnt[0];
    __syncthreads();

    // ---- exact greedy NMS ----
    // Only kept boxes suppress; classes with different ids can never overlap
    // after the reference's +2*cid offset, so cid equality is exact-equivalent.
    int i = 0;
    while (i < V) {
        int kc = s_cnt[1];                 // uniform (barrier-protected)
        if (kc >= MAXDET) break;           // later kept boxes can't reach output
        if (!s_removed[i]) {               // uniform read
            if (tid == 0) { s_kept[kc] = i; s_cnt[1] = kc + 1; }
            float4 bi = s_box[i];
            int    ci = s_cid[i];
            float areai = (bi.z - bi.x) * (bi.w - bi.y);
            for (int j = i + 1 + tid; j < V; j += 1024) {
                if (!s_removed[j] && s_cid[j] == ci) {
                    float4 bj = s_box[j];
                    float yy = fminf(bi.z, bj.z) - fmaxf(bi.x, bj.x);
                    float xx = fminf(bi.w, bj.w) - fmaxf(bi.y, bj.y);
                    float inter = fmaxf(yy, 0.f) * fmaxf(xx, 0.f);
                    float areaj = (bj.z - bj.x) * (bj.w - bj.y);
                    float iou = inter / (areai + areaj - inter + 1e-8f);
                    if (iou > NMS_TH) s_removed[j] = 1;
                }
            }
        }
        ++i;
        __syncthreads();
    }
    __syncthreads();

    // ---- emit top MAXDET (first kept in sorted order == top_k) ----
    int kc = s_cnt[1];
    if (tid < MAXDET) {
        float o0 = 0.f, o1 = 0.f, o2 = 0.f, o3 = 0.f, o4 = 0.f, o5 = 0.f;
        if (tid < kc) {
            int ii = s_kept[tid];
            float4 bx = s_box[ii];
            o0 = bx.x; o1 = bx.y; o2 = bx.z; o3 = bx.w;
            o4 = (float)s_cid[ii];
            o5 = s_score[ii];
        }
        float* row = out + ((size_t)b * MAXDET + tid) * 6;
        row[0] = o0; row[1] = o1; row[2] = o2;
        row[3] = o3; row[4] = o4; row[5] = o5;
    }
}

// -------------------------------------------------------------------------
extern "C" void kernel_launch(void* const* d_in, const int* in_sizes, int n_in,
                              void* d_out, int out_size, void* d_ws, size_t ws_size,
                              hipStream_t stream) {
    (void)in_sizes; (void)n_in; (void)out_size; (void)ws_size;
    const float* rois   = (const float*)d_in[0];
    const float* probs  = (const float*)d_in[1];
    const float* deltas = (const float*)d_in[2];
    const float* meta   = (const float*)d_in[3];

    // workspace layout: boxes (256KB) | score (64KB) | cid (64KB)
    float4* ws_box   = (float4*)d_ws;
    float*  ws_score = (float*)((char*)d_ws + sizeof(float4) * (size_t)BB * NN);
    int*    ws_cid   = (int*)(ws_score + (size_t)BB * NN);

    // 8 waves/block, one wave per ROI -> 2000 blocks
    detlayer_refine<<<(BB * NN + 7) / 8, 256, 0, stream>>>(
        rois, probs, deltas, meta, ws_box, ws_score, ws_cid);

    detlayer_sort_nms<<<BB, 1024, 0, stream>>>(
        ws_box, ws_score, ws_cid, (float*)d_out);
}